// GlacierDynamicsCheckpointed_71957882077633
// MI455X (gfx1250) — compile-verified
//
#include <hip/hip_runtime.h>
#include <hip/hip_bf16.h>
#include <stdint.h>

// ---------------- geometry / constants ----------------
constexpr int   GN     = 1024;
constexpr int   GNN    = GN * GN;
constexpr int   DROW   = 1023;               // staggered grid extent
constexpr int   NDPTS  = DROW * DROW;
constexpr int   NSTEPS = 32;
constexpr float DXF    = 200.0f;
constexpr float INV_DX = 1.0f / 200.0f;
constexpr float EPSF   = 1e-10f;
// FD * (RHO*G)^3 = 1e-17 * (910*9.81)^3
constexpr float DCOEF  = (float)(1e-17 * 8927.1 * 8927.1 * 8927.1);

// scalar slots (unsigned words) in workspace
constexpr int SC_MAXD0 = 0;    // [32]  per-step ordered max(D)
constexpr int SC_MINZ0 = 32;   // [33]  per-step ordered min(Z); slot 64 = init
constexpr int SC_FLAG0 = 65;   // [32]  smb-update flag per step
constexpr int SC_T     = 97;   // float bits: t
constexpr int SC_TLAST = 98;   // float bits: t_last

typedef __attribute__((ext_vector_type(2))) float v2f;
typedef __attribute__((ext_vector_type(8))) float v8f;

// ---------------- ordered-uint float encoding (deterministic atomics) ----
__device__ __forceinline__ unsigned f2ord(float f) {
  unsigned b = __float_as_uint(f);
  return (b & 0x80000000u) ? ~b : (b | 0x80000000u);
}
__device__ __forceinline__ float ord2f(unsigned u) {
  unsigned b = (u & 0x80000000u) ? (u & 0x7FFFFFFFu) : ~u;
  return __uint_as_float(b);
}

// ---------------- async b128 stage of a 17x65 (padded 17x68) tile --------
// GVS form: mem = SADDR(u64) + VADDR(i32); LDS dst byte addr in VGPR.
__device__ __forceinline__ void stage_tile_async(const float* __restrict__ src,
                                                 unsigned ldsBase,
                                                 int x0, int y0, int tid) {
  for (int cid = tid; cid < 17 * 17; cid += 256) {
    int r  = cid / 17;
    int ch = cid - r * 17;
    int gy = y0 + r;
    if (gy > GN - 1) gy = GN - 1;
    int lin = gy * GN + x0 + (ch << 2);          // element index, < 2^31
    if (lin > GNN - 4) lin = GNN - 4;            // keep 16B-aligned, in-bounds
    unsigned goff = (unsigned)lin << 2;          // byte offset (< 4MB)
    unsigned la   = ldsBase + (unsigned)(((r * 68) + (ch << 2)) << 2);
    asm volatile("global_load_async_to_lds_b128 %0, %1, %2"
                 :
                 : "v"(la), "v"(goff), "s"(src)
                 : "memory");
  }
}

// ---------------- kernel A: diffusivity + global max reduce --------------
// grid (16, 64), block 256. tile: 64 (x) x 16 (y) staggered points.
__global__ __launch_bounds__(256) void kA_diffusivity(
    const float* __restrict__ Hg, const float* __restrict__ Zg,
    float* __restrict__ Dg, unsigned* __restrict__ SC, int step)
{
  __shared__ float    Ht[17][68];
  __shared__ float    Zt[17][68];
  __shared__ unsigned red[256];

  const int tid = threadIdx.x;
  const int x0  = blockIdx.x * 64;
  const int y0  = blockIdx.y * 16;

  unsigned ldsH = (unsigned)(size_t)&Ht[0][0];
  unsigned ldsZ = (unsigned)(size_t)&Zt[0][0];

  stage_tile_async(Hg, ldsH, x0, y0, tid);
  stage_tile_async(Zg, ldsZ, x0, y0, tid);
  asm volatile("s_wait_asynccnt 0" ::: "memory");
  __syncthreads();

  const int tx = tid & 63;
  const int ty = tid >> 6;
  unsigned lord = 0u;

#pragma unroll
  for (int q = 0; q < 4; ++q) {
    int ly = ty + q * 4;
    int i  = y0 + ly;
    int j  = x0 + tx;
    if (i < DROW && j < DROW) {
      float h00 = Ht[ly][tx],     h01 = Ht[ly][tx + 1];
      float h10 = Ht[ly + 1][tx], h11 = Ht[ly + 1][tx + 1];
      float z00 = Zt[ly][tx],     z01 = Zt[ly][tx + 1];
      float z10 = Zt[ly + 1][tx], z11 = Zt[ly + 1][tx + 1];
      float havg = 0.25f * (h00 + h01 + h10 + h11);
      float sx = 0.5f * ((z01 - z00) + (z11 - z10)) * INV_DX;
      float sy = 0.5f * ((z10 - z00) + (z11 - z01)) * INV_DX;
      float sn = sqrtf(sx * sx + sy * sy + EPSF);
      float h2 = havg * havg;
      float d  = DCOEF * (h2 * h2 * havg) * (sn * sn) + EPSF;
      Dg[i * DROW + j] = d;
      unsigned o = f2ord(d);
      lord = (o > lord) ? o : lord;
    }
  }

  red[tid] = lord;
  __syncthreads();
  for (int w = 128; w > 0; w >>= 1) {
    if (tid < w) { unsigned o = red[tid + w]; if (o > red[tid]) red[tid] = o; }
    __syncthreads();
  }
  if (tid == 0) atomicMax(&SC[SC_MAXD0 + step], red[0]);
}

// ---------------- kernel B: dt, flux divergence, H/Z update, min(Z) ------
// grid 4096, block 256; one cell per thread; H cell-local (in-place OK),
// Z ping-pongs (zin read incl. neighbors, zout written).
__global__ __launch_bounds__(256) void kB_update(
    float* __restrict__ Hg, const float* __restrict__ zin, float* __restrict__ zout,
    const float* __restrict__ Dg, const float* __restrict__ SMB,
    const float* __restrict__ Ztopo, unsigned* __restrict__ SC, int step)
{
  __shared__ unsigned red[256];
  const int tid  = threadIdx.x;
  const int cell = blockIdx.x * 256 + tid;
  const int r = cell >> 10;
  const int c = cell & (GN - 1);

  float maxD = ord2f(SC[SC_MAXD0 + step]);
  float dt   = fminf((DXF * DXF) / (2.7f * maxD), 1.0f);

  float h  = Hg[cell];
  float hn = h;
  if (r >= 1 && r <= GN - 2 && c >= 1 && c <= GN - 2) {
    const float* D0 = Dg + (r - 1) * DROW + (c - 1);
    const float* D1 = Dg + r * DROW + (c - 1);
    float d00 = D0[0], d01 = D0[1], d10 = D1[0], d11 = D1[1];
    float zc = zin[cell];
    float zw = zin[cell - 1],  ze = zin[cell + 1];
    float zn = zin[cell - GN], zs = zin[cell + GN];
    float qxW = -0.5f * (d00 + d10) * (zc - zw) * INV_DX;
    float qxE = -0.5f * (d01 + d11) * (ze - zc) * INV_DX;
    float qyN = -0.5f * (d00 + d01) * (zc - zn) * INV_DX;
    float qyS = -0.5f * (d10 + d11) * (zs - zc) * INV_DX;
    float dHdt = -((qxE - qxW) * INV_DX + (qyS - qyN) * INV_DX);
    hn = h + dt * (dHdt + SMB[cell]);
  }
  hn = fmaxf(hn, 0.0f);
  Hg[cell] = hn;
  float zv = Ztopo[cell] + hn;
  zout[cell] = zv;

  // block-min reduce of Z, then one atomic per block
  red[tid] = f2ord(zv);
  __syncthreads();
  for (int w = 128; w > 0; w >>= 1) {
    if (tid < w) { unsigned o = red[tid + w]; if (o < red[tid]) red[tid] = o; }
    __syncthreads();
  }
  if (tid == 0) atomicMin(&SC[SC_MINZ0 + step], red[0]);

  // one thread advances simulated time and sets the smb-update flag
  if (cell == 0) {
    float* SCf = (float*)SC;
    float t  = SCf[SC_T];
    float tl = SCf[SC_TLAST];
    float tn = t + dt;
    unsigned f = ((tn - tl) >= 5.0f) ? 1u : 0u;
    SC[SC_FLAG0 + step] = f;
    SCf[SC_T] = tn;
    if (f) SCf[SC_TLAST] = tn;
  }
}

// ---------------- kernel C: surface mass balance (predicated) ------------
__global__ __launch_bounds__(256) void kC_smb(
    const float* __restrict__ Z, const float* __restrict__ PS,
    const float* __restrict__ mask, float* __restrict__ SMB,
    const float* __restrict__ pTma, const float* __restrict__ pTmj,
    const unsigned* __restrict__ SC, int flagIdx, int minIdx)
{
  if (flagIdx >= 0 && SC[flagIdx] == 0u) return;   // uniform early-out
  const int cell = blockIdx.x * 256 + threadIdx.x;

  float minZ  = ord2f(SC[minIdx]);
  float Tma0  = *pTma;
  float Tmj0  = *pTmj;
  float denom = Tmj0 - Tma0;                 // lapse cancels: T_mj - T_ma is constant
  float Tma   = Tma0 - (7.0f / 1000.0f) * (Z[cell] - minZ);

  float ratio = Tma / denom;
  float rc    = fminf(fmaxf(ratio, -1.0f + 1e-7f), 1.0f - 1e-7f);
  float ac    = acosf(rc);
  const float A = 12.0f;
  const float INVPI = 0.318309886183790672f;

  float ipt = Tma * (A - A * INVPI * ac) + denom * (A * INVPI) * sqrtf(1.0f - rc * rc);
  ipt = (ratio < 1.0f) ? ipt : 0.0f;
  float ntr = (ratio >= 1.0f) ? 0.0f : ((ratio <= -1.0f) ? 1.0f : ac * INVPI);

  // sum over 12 months: psum*ntr - 12*MELT_F*ipt ; 12*(2/12) = 2
  float smb = (PS[cell] * ntr - 2.0f * ipt) * (1000.0f / 910.0f) * mask[cell];
  SMB[cell] = smb;
}

// ---------------- init kernels ----------------
__global__ void k_init_scalars(unsigned* __restrict__ SC) {
  int i = threadIdx.x;
  if (i < 32) SC[SC_MAXD0 + i] = 0u;            // identity for ordered max
  if (i < 33) SC[SC_MINZ0 + i] = 0xFFFFFFFFu;   // identity for ordered min
  if (i < 32) SC[SC_FLAG0 + i] = 0u;
  if (i == 0) { SC[SC_T] = 0u; SC[SC_TLAST] = 0u; }
}

// 12-month precip sum as a ones-vector contraction on the f32 matrix core:
// psum(16 px) = ones(16x4) x P(4x16) accumulated over 3 K-groups.
// With A==1 and full-K accumulation the result is invariant to the K-slot
// permutation in B's register layout; only column->lane mapping matters.
__global__ __launch_bounds__(256) void k_psum(const float* __restrict__ precip,
                                              float* __restrict__ PS) {
#if __has_builtin(__builtin_amdgcn_wmma_f32_16x16x4_f32)
  const int tid  = threadIdx.x;
  const int lane = tid & 31;
  const int wav  = (blockIdx.x * 256 + tid) >> 5;   // global wave id: 0..32767
  const int col  = lane & 15;
  const int hi   = lane >> 4;                        // half-wave selects K pair
  v2f a; a.x = 1.0f; a.y = 1.0f;                     // all-ones A matrix
#pragma unroll
  for (int half = 0; half < 2; ++half) {
    const int p0 = wav * 32 + half * 16;             // 16-pixel group base
    v8f c = {};
#pragma unroll
    for (int q = 0; q < 3; ++q) {
      const int m0 = q * 4 + hi * 2;                 // months m0, m0+1 for this lane
      v2f b;
      b.x = precip[(size_t)m0 * GNN + p0 + col];
      b.y = precip[(size_t)(m0 + 1) * GNN + p0 + col];
      c = __builtin_amdgcn_wmma_f32_16x16x4_f32(false, a, false, b,
                                                (short)0, c, false, false);
    }
    if (lane < 16) PS[p0 + col] = c[0];              // D row 0 = column sums
  }
#else
  int i = blockIdx.x * 256 + threadIdx.x;
  float s = 0.0f;
#pragma unroll
  for (int m = 0; m < 12; ++m) s += precip[(size_t)m * GNN + i];
  PS[i] = s;
#endif
}

__global__ __launch_bounds__(256) void k_init_fields(
    const float* __restrict__ Ztopo, float* __restrict__ Hg,
    float* __restrict__ Za, unsigned* __restrict__ SC)
{
  __shared__ unsigned red[256];
  const int tid  = threadIdx.x;
  const int cell = blockIdx.x * 256 + tid;
  Hg[cell] = 0.0f;
  float z = Ztopo[cell];
  Za[cell] = z;
  red[tid] = f2ord(z);
  __syncthreads();
  for (int w = 128; w > 0; w >>= 1) {
    if (tid < w) { unsigned o = red[tid + w]; if (o < red[tid]) red[tid] = o; }
    __syncthreads();
  }
  if (tid == 0) atomicMin(&SC[SC_MINZ0 + 32], red[0]);  // init slot (64)
}

// ---------------- launcher ----------------
extern "C" void kernel_launch(void* const* d_in, const int* in_sizes, int n_in,
                              void* d_out, int out_size, void* d_ws, size_t ws_size,
                              hipStream_t stream) {
  (void)in_sizes; (void)n_in; (void)out_size; (void)ws_size;

  const float* Ztopo  = (const float*)d_in[0];
  const float* mask   = (const float*)d_in[1];
  const float* precip = (const float*)d_in[2];
  const float* pTma   = (const float*)d_in[3];
  const float* pTmj   = (const float*)d_in[4];
  // d_in[5] = unroll (BPTT knob) — irrelevant for forward result

  float* H  = (float*)d_out;          // H lives in d_out the whole run
  float* ws = (float*)d_ws;
  float* Za  = ws;                    // Z ping buffer
  float* Zb  = ws + (size_t)GNN;      // Z pong buffer
  float* SMB = ws + (size_t)2 * GNN;
  float* PS  = ws + (size_t)3 * GNN;
  float* Dg  = ws + (size_t)4 * GNN;
  unsigned* SC = (unsigned*)(ws + (size_t)4 * GNN + (size_t)NDPTS);

  const int nb = GNN / 256;           // 4096 blocks

  k_init_scalars<<<1, 128, 0, stream>>>(SC);
  k_psum<<<nb, 256, 0, stream>>>(precip, PS);
  k_init_fields<<<nb, 256, 0, stream>>>(Ztopo, H, Za, SC);
  // initial smb from Z0 (unconditional; min(Z0) in slot 64)
  kC_smb<<<nb, 256, 0, stream>>>(Za, PS, mask, SMB, pTma, pTmj, SC, -1, SC_MINZ0 + 32);

  for (int s = 0; s < NSTEPS; ++s) {
    float* zin  = (s & 1) ? Zb : Za;
    float* zout = (s & 1) ? Za : Zb;
    kA_diffusivity<<<dim3(16, 64), 256, 0, stream>>>(H, zin, Dg, SC, s);
    kB_update<<<nb, 256, 0, stream>>>(H, zin, zout, Dg, SMB, Ztopo, SC, s);
    kC_smb<<<nb, 256, 0, stream>>>(zout, PS, mask, SMB, pTma, pTmj, SC,
                                   SC_FLAG0 + s, SC_MINZ0 + s);
  }
}